// ConstrastiveCrossViewLucasVSCorineLoss_89051851915601
// MI455X (gfx1250) — compile-verified
//
#include <hip/hip_runtime.h>
#include <hip/hip_bf16.h>
#include <math.h>

// ---------------------------------------------------------------------------
// Problem constants (from reference): B=4, D=768, h=w=128, labels 512x512,
// N = B*h*w = 65536, C = 9 classes. Output: single f32 scalar.
// ---------------------------------------------------------------------------
#define NC        9
#define DIM       768
#define PLANE     16384      // h*w = 128*128
#define NTOT      65536      // B*PLANE
#define LBSTRIDE  262144     // 512*512 per batch
#define MPAD      16         // M padded 9 -> 16 for WMMA

// Workspace layout (in floats)
#define SUMS_OFF    0        // [9*768]
#define COUNTS_OFF  6912     // [9]
#define ROWN_OFF    6928     // [16]
#define LOSS_OFF    6944     // [1]
#define APAD_OFF    7168     // [16*768]
#define LOGITS_OFF  19456    // [9*65536]
#define ZERO_N      6960     // floats to zero each call

typedef float v2f __attribute__((ext_vector_type(2)));
typedef float v8f __attribute__((ext_vector_type(8)));

// ---------------------------------------------------------------------------
// K0: zero the small accumulators (sums, counts, rownorm2, lossAcc)
// ---------------------------------------------------------------------------
__global__ __launch_bounds__(256) void k0_zero(float* __restrict__ ws) {
  int i = blockIdx.x * 256 + threadIdx.x;
  if (i < ZERO_N) ws[i] = 0.0f;
}

// ---------------------------------------------------------------------------
// K1: per-class segment sums.  One wave per (b, d) channel plane (3072 waves).
// Lane walks the 16384-pixel plane with stride 32 (coalesced), keeps 9
// register accumulators, cross-lane reduces, lane 0 does 9 global atomics.
// ---------------------------------------------------------------------------
__global__ __launch_bounds__(256) void k1_segsum(const float* __restrict__ feat,
                                                 const int*   __restrict__ label,
                                                 float*       __restrict__ sums) {
  const int gw   = (blockIdx.x * 256 + threadIdx.x) >> 5;   // 0..3071
  const int lane = threadIdx.x & 31;
  const int b    = gw / DIM;
  const int d    = gw - b * DIM;
  const float* base = feat + (size_t)(b * DIM + d) * PLANE;
  const int*   lb   = label + (size_t)b * LBSTRIDE;

  float acc[NC];
#pragma unroll
  for (int c = 0; c < NC; ++c) acc[c] = 0.0f;

  for (int p = lane; p < PLANE; p += 32) {
    // corine[b,i,j] = label[b, 4*i, 4*j]; i = p>>7, j = p&127
    const int cls = lb[((p >> 7) << 11) + ((p & 127) << 2)];
    const float v = base[p];
#pragma unroll
    for (int c = 0; c < NC; ++c) acc[c] += (cls == c) ? v : 0.0f;
  }

#pragma unroll
  for (int c = 0; c < NC; ++c) {
    float r = acc[c];
    r += __shfl_xor(r, 1, 32);
    r += __shfl_xor(r, 2, 32);
    r += __shfl_xor(r, 4, 32);
    r += __shfl_xor(r, 8, 32);
    r += __shfl_xor(r, 16, 32);
    if (lane == 0) atomicAdd(&sums[c * DIM + d], r);
  }
}

// ---------------------------------------------------------------------------
// K1b: per-class pixel counts via LDS histogram.
// ---------------------------------------------------------------------------
__global__ __launch_bounds__(256) void k1b_counts(const int* __restrict__ label,
                                                  float*     __restrict__ counts) {
  __shared__ float hist[NC];
  if (threadIdx.x < NC) hist[threadIdx.x] = 0.0f;
  __syncthreads();
  const int n   = blockIdx.x * 256 + threadIdx.x;      // 0..65535
  const int b   = n >> 14;
  const int rem = n & (PLANE - 1);
  const int cls = label[(size_t)b * LBSTRIDE + ((rem >> 7) << 11) + ((rem & 127) << 2)];
  atomicAdd(&hist[cls], 1.0f);
  __syncthreads();
  if (threadIdx.x < NC) atomicAdd(&counts[threadIdx.x], hist[threadIdx.x]);
}

// ---------------------------------------------------------------------------
// K2: protos = normalize(0.99*sums/counts + 0.01*prototypes), write padded
// 16x768 A-matrix (rows 9..15 zero) for the WMMA GEMM.
// One block per padded row (grid = 16).
// ---------------------------------------------------------------------------
__global__ __launch_bounds__(256) void k2_protos(const float* __restrict__ sums,
                                                 const float* __restrict__ counts,
                                                 const float* __restrict__ protoIn,
                                                 float*       __restrict__ Apad) {
  const int c = blockIdx.x;           // 0..15
  const int t = threadIdx.x;
  if (c >= NC) {
    for (int d = t; d < DIM; d += 256) Apad[c * DIM + d] = 0.0f;
    return;
  }
  __shared__ float red[256];
  const float invCnt = 1.0f / counts[c];
  float vals[3];
  float ss = 0.0f;
#pragma unroll
  for (int r = 0; r < 3; ++r) {
    const int d = t + r * 256;
    const float p = 0.99f * (sums[c * DIM + d] * invCnt) + 0.01f * protoIn[c * DIM + d];
    vals[r] = p;
    ss += p * p;
  }
  red[t] = ss;
  __syncthreads();
  for (int s = 128; s > 0; s >>= 1) {
    if (t < s) red[t] += red[t + s];
    __syncthreads();
  }
  const float inv = 1.0f / fmaxf(sqrtf(red[0]), 1e-12f);
#pragma unroll
  for (int r = 0; r < 3; ++r) Apad[c * DIM + t + r * 256] = vals[r] * inv;
}

// ---------------------------------------------------------------------------
// K3: logits = Apad(16x768) @ features[b](768x16384), per batch plane.
// One 16x16 output tile per wave (4096 tiles), K swept in steps of 4 with
// v_wmma_f32_16x16x4_f32.  A staged in LDS (48 KB of the 320 KB WGP LDS).
// Also accumulates per-row sum-of-squares for the row L2 norm over N.
//
// VGPR layouts (ISA 7.12.2):
//   A 16x4 f32 : lane m(0..15) v0=K(khalf), v1=K(khalf+1), khalf = (lane>>4)*2
//   B 4x16 f32 : lane n(0..15) v0=K(khalf), v1=K(khalf+1)
//   C/D 16x16  : vgpr j, lanes 0-15 -> M=j, lanes 16-31 -> M=8+j, N=lane&15
// ---------------------------------------------------------------------------
__global__ __launch_bounds__(256) void k3_gemm(const float* __restrict__ feat,
                                               const float* __restrict__ Apad,
                                               float*       __restrict__ logits,
                                               float*       __restrict__ rownorm2) {
  __shared__ float lA[MPAD * DIM];
  for (int i = threadIdx.x; i < MPAD * DIM; i += 256) lA[i] = Apad[i];
  __syncthreads();

  const int lane  = threadIdx.x & 31;
  const int wave  = threadIdx.x >> 5;
  const int tile  = blockIdx.x * 8 + wave;        // 0..4095
  const int n0    = tile * 16;                    // global column base
  const int b     = n0 >> 14;
  const int nInB  = n0 & (PLANE - 1);
  const float* plane = feat + (size_t)b * (DIM * PLANE);

  const int col   = lane & 15;
  const int khalf = (lane >> 4) << 1;             // 0 or 2
  const float* arow = &lA[col * DIM];

  v8f c = {0.f, 0.f, 0.f, 0.f, 0.f, 0.f, 0.f, 0.f};
  for (int k0 = 0; k0 < DIM; k0 += 4) {
    v2f av, bv;
    av.x = arow[k0 + khalf];
    av.y = arow[k0 + khalf + 1];
    const float* bp = plane + (size_t)(k0 + khalf) * PLANE + nInB + col;
    bv.x = bp[0];
    bv.y = bp[PLANE];
    c = __builtin_amdgcn_wmma_f32_16x16x4_f32(false, av, false, bv,
                                              (short)0, c, false, false);
  }

  const int mbase = (lane >> 4) << 3;             // 0 or 8
#pragma unroll
  for (int j = 0; j < 8; ++j) {
    const int m = mbase + j;
    const float v = c[j];
    if (m < NC) logits[(size_t)m * NTOT + n0 + col] = v;
    float sq = v * v;                              // reduce over the 16-lane half
    sq += __shfl_xor(sq, 1, 16);
    sq += __shfl_xor(sq, 2, 16);
    sq += __shfl_xor(sq, 4, 16);
    sq += __shfl_xor(sq, 8, 16);
    if (col == 0 && m < NC) atomicAdd(&rownorm2[m], sq);
  }
}

// ---------------------------------------------------------------------------
// K4: per-pixel cross-entropy term; block tree-reduce; atomic into lossAcc.
// prot_feat[c,n] = logits[c,n]/max(||row c||,eps) / 0.1 ; row 2 := (lc==2)
// loss_n = log(sum_c exp(pf)) - pf[lc]
// ---------------------------------------------------------------------------
__global__ __launch_bounds__(256) void k4_loss(const float* __restrict__ logits,
                                               const float* __restrict__ rownorm2,
                                               const int*   __restrict__ label,
                                               float*       __restrict__ lossAcc) {
  __shared__ float red[256];
  const int n = blockIdx.x * 256 + threadIdx.x;   // 0..65535

  float inv[NC];
#pragma unroll
  for (int c = 0; c < NC; ++c)
    inv[c] = 10.0f / fmaxf(sqrtf(rownorm2[c]), 1e-12f);   // 1/(norm*TEMP)

  const int b   = n >> 14;
  const int rem = n & (PLANE - 1);
  const int raw = label[(size_t)b * LBSTRIDE + ((rem >> 7) << 11) + ((rem & 127) << 2)];
  const int lc  = (raw == 7) ? 6 : raw;

  float pf[NC];
#pragma unroll
  for (int c = 0; c < NC; ++c)
    pf[c] = (c == 2) ? ((lc == 2) ? 1.0f : 0.0f)
                     : logits[(size_t)c * NTOT + n] * inv[c];

  float A2 = 0.0f, pl = 0.0f;
#pragma unroll
  for (int c = 0; c < NC; ++c) {
    A2 += expf(pf[c]);
    pl = (c == lc) ? pf[c] : pl;
  }
  const float loss = logf(A2) - pl;

  red[threadIdx.x] = loss;
  __syncthreads();
  for (int s = 128; s > 0; s >>= 1) {
    if (threadIdx.x < s) red[threadIdx.x] += red[threadIdx.x + s];
    __syncthreads();
  }
  if (threadIdx.x == 0) atomicAdd(lossAcc, red[0]);
}

// ---------------------------------------------------------------------------
// K5: scalar finalize
// ---------------------------------------------------------------------------
__global__ void k5_final(const float* __restrict__ lossAcc, float* __restrict__ out) {
  if (threadIdx.x == 0 && blockIdx.x == 0) out[0] = lossAcc[0] * (1.0f / (float)NTOT);
}

// ---------------------------------------------------------------------------
// Inputs (setup_inputs order): 0=cls_score(unused), 1=label, 2=gt_lucas(unused),
// 3=features, 4=prototypes.  Output: 1 float.
// ---------------------------------------------------------------------------
extern "C" void kernel_launch(void* const* d_in, const int* in_sizes, int n_in,
                              void* d_out, int out_size, void* d_ws, size_t ws_size,
                              hipStream_t stream) {
  (void)in_sizes; (void)n_in; (void)out_size; (void)ws_size;
  const int*   label   = (const int*)d_in[1];
  const float* feat    = (const float*)d_in[3];
  const float* protoIn = (const float*)d_in[4];
  float* out = (float*)d_out;
  float* ws  = (float*)d_ws;

  float* sums     = ws + SUMS_OFF;
  float* counts   = ws + COUNTS_OFF;
  float* rownorm2 = ws + ROWN_OFF;
  float* lossAcc  = ws + LOSS_OFF;
  float* Apad     = ws + APAD_OFF;
  float* logits   = ws + LOGITS_OFF;

  k0_zero   <<<(ZERO_N + 255) / 256, 256, 0, stream>>>(ws);
  k1_segsum <<<384, 256, 0, stream>>>(feat, label, sums);        // 3072 waves
  k1b_counts<<<NTOT / 256, 256, 0, stream>>>(label, counts);
  k2_protos <<<MPAD, 256, 0, stream>>>(sums, counts, protoIn, Apad);
  k3_gemm   <<<512, 256, 0, stream>>>(feat, Apad, logits, rownorm2);  // 4096 tiles
  k4_loss   <<<NTOT / 256, 256, 0, stream>>>(logits, rownorm2, label, lossAcc);
  k5_final  <<<1, 1, 0, stream>>>(lossAcc, out);
}